// EmbeddingVariable_28355374088862
// MI455X (gfx1250) — compile-verified
//
#include <hip/hip_runtime.h>
#include <stdint.h>

// Embedding gather: out[r, :] = table[ids[r], :] for r in [0, 204800), D=64 f32.
// Pure bandwidth problem (~100 MB -> ~4.3 us @ 23.3 TB/s). Fully DMA-offloaded
// on CDNA5: TDM GATHER descriptors pull 8 random table rows each into LDS
// (32-bit row indices), then one TDM STORE descriptor streams the contiguous
// 8 KB LDS tile to the output slice. The wave itself executes only scalar code.

#define ROWS            (4096 * 50)     // 204800
#define DIM             64              // floats per row (256 B)
#define VOCAB           1000000u
#define ROWS_PER_DESC   8               // 32-bit gather indices -> max 8 rows/D#
#define DESC_PER_BLOCK  4
#define ROWS_PER_BLOCK  (ROWS_PER_DESC * DESC_PER_BLOCK)   // 32
#define TILE_BYTES      (ROWS_PER_BLOCK * DIM * 4)         // 8192
#define TILE_ELEMS      (ROWS_PER_BLOCK * DIM)             // 2048 x f32
#define NBLOCKS         (ROWS / ROWS_PER_BLOCK)            // 6400 (exact)

typedef uint32_t u32;
typedef uint64_t u64;
typedef u32 v4u __attribute__((ext_vector_type(4)));
typedef u32 v8u __attribute__((ext_vector_type(8)));

__global__ __launch_bounds__(32)
void EmbeddingVariable_tdm_gather(const int* __restrict__ ids,
                                  const float* __restrict__ table,
                                  float* __restrict__ out) {
  // 32 rows * 64 f32 = 8 KB staging tile; rows packed back-to-back so the tile
  // is an exact image of the contiguous output slice out[rowBase..rowBase+32).
  __shared__ float4 smem[TILE_ELEMS / 4];

  const u32 rowBase = blockIdx.x * ROWS_PER_BLOCK;          // uniform

  const u64 tbl     = (u64)(uintptr_t)table;                // table[0][0] byte addr
  const u64 dst     = (u64)(uintptr_t)out + (u64)rowBase * (DIM * 4);
  // Generic LDS address: low 32 bits are the wave-relative LDS byte offset
  // (flat-aperture rule, ISA 10.2) — exactly what D#.lds_addr wants.
  const u32 ldsBase = (u32)(uintptr_t)(void*)smem;

  // ==== GATHER-LOAD descriptors (4 x 8 rows) ===============================
  // D# group 1 (256 bits, constant across the 4 loads):
  //   [17:16] data_size=2 (4 B)      [79:48]  tensor_dim0 = 64
  //   [111:80] tensor_dim1 = 1e6     [127:112] tile_dim0 = 64
  //   [143:128] tile_dim1 = 8 (#valid 32-bit indices)
  //   [207:160] tensor_dim0_stride = 64 elements (dim1 stride ignored: gather)
  v8u g1;
  g1[0] = 2u << 16;
  g1[1] = ((u32)DIM & 0xFFFFu) << 16;
  g1[2] = (((u32)DIM >> 16) & 0xFFFFu) | ((VOCAB & 0xFFFFu) << 16);
  g1[3] = (VOCAB >> 16) | ((u32)DIM << 16);
  g1[4] = ROWS_PER_DESC;
  g1[5] = (u32)DIM;
  g1[6] = 0;
  g1[7] = 0;

#pragma unroll
  for (int d = 0; d < DESC_PER_BLOCK; ++d) {
    const u32 r0 = rowBase + (u32)d * ROWS_PER_DESC;        // uniform -> s_load

    // D# group 0: [1:0] count=1, [30] 32-bit indices, [31] gather_mode=1,
    // [63:32] lds_addr, [120:64] global base of table, [127:126] type=2.
    v4u g0;
    g0[0] = 0xC0000001u;
    g0[1] = ldsBase + (u32)d * (ROWS_PER_DESC * DIM * 4);
    g0[2] = (u32)tbl;
    g0[3] = ((u32)(tbl >> 32) & 0x01FFFFFFu) | (2u << 30);

    // Groups 2/3: eight 32-bit row indices (all < VOCAB, so the gather-mode
    // OOB monotonicity caveat never triggers).
    v4u g2, g3;
    g2[0] = (u32)ids[r0 + 0]; g2[1] = (u32)ids[r0 + 1];
    g2[2] = (u32)ids[r0 + 2]; g2[3] = (u32)ids[r0 + 3];
    g3[0] = (u32)ids[r0 + 4]; g3[1] = (u32)ids[r0 + 5];
    g3[2] = (u32)ids[r0 + 6]; g3[3] = (u32)ids[r0 + 7];

    asm volatile("tensor_load_to_lds %0, %1, %2, %3"
                 :: "s"(g0), "s"(g1), "s"(g2), "s"(g3)
                 : "memory");
  }

  // All 4 gather tiles resident in LDS before the TDM store may read them.
  asm volatile("s_wait_tensorcnt 0x0" ::: "memory");

  // ==== STORE descriptor: 1-D, 2048 contiguous f32 (8 KB) LDS -> out =======
  {
    v4u s0;
    s0[0] = 1u;                                             // count=1, no gather
    s0[1] = ldsBase;
    s0[2] = (u32)dst;
    s0[3] = ((u32)(dst >> 32) & 0x01FFFFFFu) | (2u << 30);  // addr hi | type=2

    v8u s1;
    s1[0] = 2u << 16;                                       // data_size = 4 B
    s1[1] = ((u32)TILE_ELEMS & 0xFFFFu) << 16;              // tensor_dim0 lo16
    s1[2] = (((u32)TILE_ELEMS >> 16) & 0xFFFFu)             // tensor_dim0 hi16
          | (1u << 16);                                     // tensor_dim1 = 1
    s1[3] = 0u | ((u32)TILE_ELEMS << 16);                   // tile_dim0 = 2048
    s1[4] = 0;                                              // tile_dim1/2 unused (1-D)
    s1[5] = (u32)TILE_ELEMS;                                // tensor_dim0_stride
    s1[6] = 0;
    s1[7] = 0;

    const v4u z = {0, 0, 0, 0};                             // dims 3/4 unused
    asm volatile("tensor_store_from_lds %0, %1, %2, %3"
                 :: "s"(s0), "s"(s1), "s"(z), "s"(z)
                 : "memory");
  }

  // S_ENDPGM does an implicit wait-idle, but be explicit: the TDM must finish
  // reading LDS before this workgroup's allocation is released.
  asm volatile("s_wait_tensorcnt 0x0" ::: "memory");
}

extern "C" void kernel_launch(void* const* d_in, const int* in_sizes, int n_in,
                              void* d_out, int out_size, void* d_ws, size_t ws_size,
                              hipStream_t stream) {
  const int*   ids   = (const int*)d_in[0];    // [4096*50] int32
  const float* table = (const float*)d_in[1];  // [1e6 * 64] float32
  float*       out   = (float*)d_out;          // [4096*50*64] float32

  (void)in_sizes; (void)n_in; (void)out_size; (void)d_ws; (void)ws_size;

  EmbeddingVariable_tdm_gather<<<NBLOCKS, 32, 0, stream>>>(ids, table, out);
}